// DynamicBackdoorGenerator_82712480186424
// MI455X (gfx1250) — compile-verified
//
#include <hip/hip_runtime.h>
#include <hip/hip_bf16.h>
#include <math.h>

#define Bsz 32
#define Ssz 128
#define Vsz 30522
#define Hsz 768
#define Tsz 16
#define PAD_ID 0
#define SEP_ID 102
#define MASK_ID 103
#define TEMP 0.48f
#define NEG_INF -1e9f
#define EPSG 1e-10f
#define SCALE 0.03608439182435161f  /* 1/sqrt(768) */

typedef float v2f __attribute__((ext_vector_type(2)));
typedef float v8f __attribute__((ext_vector_type(8)));
typedef unsigned int u32x4 __attribute__((ext_vector_type(4)));
typedef int i32x4 __attribute__((ext_vector_type(4)));
typedef int i32x8 __attribute__((ext_vector_type(8)));

#if defined(__has_builtin)
#if __has_builtin(__builtin_amdgcn_tensor_load_to_lds)
#define HAVE_TDM 1
#endif
#endif

// ---------------- init kernels ----------------

__global__ void k_zero(float* p, long n) {
  long i = (long)blockIdx.x * blockDim.x + threadIdx.x;
  if (i < n) p[i] = 0.0f;
}

__global__ void k_init_state(const int* eos_location, int* eos, int* done) {
  int b = threadIdx.x;
  if (b < Bsz) { eos[b] = eos_location[b]; done[b] = 0; }
}

__global__ void k_bias(const int* input_ids, float* bias) {
  int i = blockIdx.x * blockDim.x + threadIdx.x;
  if (i < Bsz * Ssz) bias[i] = (input_ids[i] != PAD_ID) ? 0.0f : NEG_INF;
}

__global__ void k_gather_embeds(const int* input_ids, const float* E, float* embeds) {
  long i = (long)blockIdx.x * blockDim.x + threadIdx.x;
  if (i < (long)Bsz * Ssz * Hsz) {
    int pos = (int)(i / Hsz);
    int h = (int)(i - (long)pos * Hsz);
    embeds[i] = E[(size_t)input_ids[pos] * Hsz + h];
  }
}

// mk = mask_embed@Wk, mv = mask_embed@Wv, mq = mask_embed@Wq (step-invariant)
__global__ void k_mask_rows(const float* E, const float* Wk, const float* Wv,
                            const float* Wq, float* mk, float* mv, float* mq) {
  int j = blockIdx.x * blockDim.x + threadIdx.x;
  if (j >= 3 * Hsz) return;
  int mat = j / Hsz, h = j - mat * Hsz;
  const float* W = (mat == 0) ? Wk : (mat == 1) ? Wv : Wq;
  const float* me = E + (size_t)MASK_ID * Hsz;
  float d = 0.0f;
  for (int i = 0; i < Hsz; ++i) d = fmaf(me[i], W[(size_t)i * Hsz + h], d);
  float* dst = (mat == 0) ? mk : (mat == 1) ? mv : mq;
  dst[h] = d;
}

// ---------------- WMMA GEMM: C[M,768] = A[M,768] @ Bm[768,768] ----------------
// grid (M/32, 48), 64 threads = 2 waves; wave w handles M-tile m0+16w.
// Two independent accumulator chains over split-K for WMMA ILP.
__global__ void __launch_bounds__(64)
k_gemm_nn_wmma(const float* __restrict__ A, const float* __restrict__ Bm,
               float* __restrict__ C) {
  int tid = threadIdx.x;
  int wave = tid >> 5, lane = tid & 31;
  int m0 = blockIdx.x * 32 + wave * 16;
  int n0 = blockIdx.y * 16;
  int m = lane & 15, n = lane & 15;
  int kb = (lane >> 4) << 1;                 // 0 for lanes 0-15, 2 for lanes 16-31
  const float* arow = A + (size_t)(m0 + m) * Hsz + kb;
  v8f acc0 = {}, acc1 = {};
  for (int k0 = 0; k0 < Hsz; k0 += 8) {
    v2f a0, b0, a1, b1;
    a0.x = arow[k0];
    a0.y = arow[k0 + 1];
    b0.x = Bm[(size_t)(k0 + kb) * Hsz + n0 + n];
    b0.y = Bm[(size_t)(k0 + kb + 1) * Hsz + n0 + n];
    a1.x = arow[k0 + 4];
    a1.y = arow[k0 + 5];
    b1.x = Bm[(size_t)(k0 + 4 + kb) * Hsz + n0 + n];
    b1.y = Bm[(size_t)(k0 + 5 + kb) * Hsz + n0 + n];
    acc0 = __builtin_amdgcn_wmma_f32_16x16x4_f32(false, a0, false, b0,
                                                 (short)0, acc0, false, false);
    acc1 = __builtin_amdgcn_wmma_f32_16x16x4_f32(false, a1, false, b1,
                                                 (short)0, acc1, false, false);
  }
  int rbase = m0 + ((lane >> 4) << 3);       // M = r + 8*(lane>=16) within tile
  for (int r = 0; r < 8; ++r)
    C[(size_t)(rbase + r) * Hsz + n0 + n] = acc0[r] + acc1[r];
}

// ---------------- per-step kernels ----------------

__global__ void k_scatter_mask(float* embeds, float* Kc, float* Vc, const float* E,
                               const float* mk, const float* mv, const int* eos) {
  int b = blockIdx.x;
  int e = eos[b];
  const float* me = E + (size_t)MASK_ID * Hsz;
  size_t base = ((size_t)b * Ssz + e) * Hsz;
  for (int h = threadIdx.x; h < Hsz; h += blockDim.x) {
    embeds[base + h] = me[h];
    Kc[base + h] = mk[h];
    Vc[base + h] = mv[h];
  }
}

// one block per batch: scores(eos row) -> softmax -> ctx -> layernorm -> h_eos
__global__ void __launch_bounds__(256)
k_attention(const float* embeds, const float* Kc, const float* Vc,
            const float* bias, const float* mq, const float* gamma,
            const float* beta, const int* eos, float* h_eos) {
  __shared__ float sq[Hsz];
  __shared__ float sa[Ssz];
  __shared__ float red[256];
  __shared__ float sx[Hsz];
  int b = blockIdx.x, tid = threadIdx.x;
  int e = eos[b];
  for (int h = tid; h < Hsz; h += 256) sq[h] = mq[h];
  __syncthreads();

  float sc = -INFINITY;
  if (tid < Ssz) {
    const float* krow = Kc + ((size_t)b * Ssz + tid) * Hsz;
    float d = 0.0f;
    for (int h = 0; h < Hsz; ++h) d = fmaf(sq[h], krow[h], d);
    sc = d * SCALE + bias[b * Ssz + tid];
    sa[tid] = sc;
  }
  red[tid] = sc;
  __syncthreads();
  for (int s = 128; s > 0; s >>= 1) {
    if (tid < s) red[tid] = fmaxf(red[tid], red[tid + s]);
    __syncthreads();
  }
  float mx = red[0];
  __syncthreads();
  float ex = 0.0f;
  if (tid < Ssz) { ex = __expf(sa[tid] - mx); }
  red[tid] = ex;
  __syncthreads();
  for (int s = 128; s > 0; s >>= 1) {
    if (tid < s) red[tid] += red[tid + s];
    __syncthreads();
  }
  float inv = 1.0f / red[0];
  __syncthreads();
  if (tid < Ssz) sa[tid] = ex * inv;
  __syncthreads();

  size_t ebase = ((size_t)b * Ssz + e) * Hsz;
  for (int h = tid; h < Hsz; h += 256) {
    float c = 0.0f;
    for (int s = 0; s < Ssz; ++s)
      c = fmaf(sa[s], Vc[((size_t)b * Ssz + s) * Hsz + h], c);
    sx[h] = embeds[ebase + h] + c;
  }
  __syncthreads();
  // layernorm over H
  float ps = 0.0f;
  for (int h = tid; h < Hsz; h += 256) ps += sx[h];
  red[tid] = ps;
  __syncthreads();
  for (int s = 128; s > 0; s >>= 1) {
    if (tid < s) red[tid] += red[tid + s];
    __syncthreads();
  }
  float mu = red[0] * (1.0f / Hsz);
  __syncthreads();
  float pv = 0.0f;
  for (int h = tid; h < Hsz; h += 256) { float d = sx[h] - mu; pv = fmaf(d, d, pv); }
  red[tid] = pv;
  __syncthreads();
  for (int s = 128; s > 0; s >>= 1) {
    if (tid < s) red[tid] += red[tid + s];
    __syncthreads();
  }
  float rs = rsqrtf(red[0] * (1.0f / Hsz) + 1e-12f);
  for (int h = tid; h < Hsz; h += 256)
    h_eos[(size_t)b * Hsz + h] = (sx[h] - mu) * rs * gamma[h] + beta[h];
}

// logits[32, V] = h_eos[32,768] @ E^T ; one 16-wide N tile per block, 2 waves = 2 M tiles.
// E tile staged into LDS by the Tensor Data Mover with 1-DWORD padding every 256 DWORDs:
// effective row stride = 771 floats (771 mod 64 = 3, gcd(3,64)=1 -> conflict-free
// transposed reads). Element (r,c) lives at r*771 + c + (c>>8).
#define LROW 771
__global__ void __launch_bounds__(64)
k_logits_wmma(const float* __restrict__ hE, const float* __restrict__ E,
              float* __restrict__ logits) {
  __shared__ float shE[16 * LROW];
  int n0 = blockIdx.x * 16;
  int tid = threadIdx.x;
#ifdef HAVE_TDM
  if (tid < 32) {  // wave 0 issues the TDM descriptor (EXEC-independent, one issue)
    unsigned lds_addr = (unsigned)(uintptr_t)(&shE[0]);
    unsigned long long ga = (unsigned long long)(uintptr_t)(E + (size_t)n0 * Hsz);
    unsigned dim1 = (unsigned)(Vsz - n0);    // TDM OOB rows read as zero (ragged tail)
    u32x4 g0;
    g0[0] = 1u;                               // count=1, user descriptor
    g0[1] = lds_addr;                         // lds_addr
    g0[2] = (unsigned)(ga & 0xFFFFFFFFu);     // global_addr[31:0]
    g0[3] = (unsigned)((ga >> 32) & 0x1FFFFFFu) | (2u << 30);  // addr[56:32] | type=2
    i32x8 g1;
    g1[0] = (int)((2u << 16) | (1u << 20) | (7u << 22)); // data=4B, pad_en, every 256DW, +1DW
    g1[1] = (int)(768u << 16);                // tensor_dim0[15:0] @ bits63:48
    g1[2] = (int)((dim1 & 0xFFFFu) << 16);    // dim0 hi=0 | tensor_dim1[15:0]
    g1[3] = (int)(768u << 16);                // dim1 hi=0 | tile_dim0=768
    g1[4] = 16;                               // tile_dim1=16, tile_dim2=0
    g1[5] = 768;                              // tensor_dim0_stride low
    g1[6] = 0;                                // stride hi | dim1_stride low
    g1[7] = 0;
    i32x4 gz = {};
#if defined(__clang_major__) && (__clang_major__ >= 23)
    i32x8 gz8 = {};
    __builtin_amdgcn_tensor_load_to_lds(g0, g1, gz, gz, gz8, 0);
#else
    __builtin_amdgcn_tensor_load_to_lds(g0, g1, gz, gz, 0);
#endif
#if __has_builtin(__builtin_amdgcn_s_wait_tensorcnt)
    __builtin_amdgcn_s_wait_tensorcnt(0);
#else
    asm volatile("s_wait_tensorcnt 0x0" ::: "memory");
#endif
  }
  __syncthreads();
#else
  for (int i = tid; i < 16 * Hsz; i += 64) {
    int r = i / Hsz, c = i - r * Hsz;
    int row = n0 + r;
    shE[r * LROW + c + (c >> 8)] = (row < Vsz) ? E[(size_t)row * Hsz + c] : 0.0f;
  }
  __syncthreads();
#endif
  int wave = tid >> 5, lane = tid & 31;
  int m = lane & 15, n = lane & 15;
  int kb = (lane >> 4) << 1;
  const float* arow = hE + (size_t)(wave * 16 + m) * Hsz + kb;
  const float* brow = shE + n * LROW + kb;
  v8f acc0 = {}, acc1 = {};
  for (int k0 = 0; k0 < Hsz; k0 += 8) {
    int p0 = k0 >> 8;             // pad DWORDs inserted before column k0
    int p1 = (k0 + 4) >> 8;
    v2f a0, b0, a1, b1;
    a0.x = arow[k0];
    a0.y = arow[k0 + 1];
    b0.x = brow[k0 + p0];
    b0.y = brow[k0 + p0 + 1];
    a1.x = arow[k0 + 4];
    a1.y = arow[k0 + 5];
    b1.x = brow[k0 + 4 + p1];
    b1.y = brow[k0 + 5 + p1];
    acc0 = __builtin_amdgcn_wmma_f32_16x16x4_f32(false, a0, false, b0,
                                                 (short)0, acc0, false, false);
    acc1 = __builtin_amdgcn_wmma_f32_16x16x4_f32(false, a1, false, b1,
                                                 (short)0, acc1, false, false);
  }
  int rbase = wave * 16 + ((lane >> 4) << 3);
  int nn = n0 + n;
  if (nn < Vsz)
    for (int r = 0; r < 8; ++r)
      logits[(size_t)(rbase + r) * Vsz + nn] = acc0[r] + acc1[r];
}

// online softmax + argmax over V; write sparse straight-through output
__global__ void __launch_bounds__(256)
k_gumbel_argmax(const float* __restrict__ logits, const float* __restrict__ u_noise,
                int t, const int* done, int* idx_out, float* out) {
  __shared__ float sm[256], ss[256];
  __shared__ int si[256];
  int b = blockIdx.x, tid = threadIdx.x;
  const float* lrow = logits + (size_t)b * Vsz;
  const float* urow = u_noise + ((size_t)t * Bsz + b) * Vsz;
  float m = -INFINITY, s = 0.0f;
  int am = 0;
  for (int v = tid; v < Vsz; v += 256) {
    float g = -logf(-logf(urow[v] + EPSG) + EPSG);
    float z = (lrow[v] + g) * (1.0f / TEMP);
    if (z > m) { s = s * __expf(m - z) + 1.0f; am = v; m = z; }
    else       { s += __expf(z - m); }
  }
  sm[tid] = m; ss[tid] = s; si[tid] = am;
  __syncthreads();
  for (int w = 128; w > 0; w >>= 1) {
    if (tid < w) {
      float m1 = sm[tid], s1 = ss[tid];
      float m2 = sm[tid + w], s2 = ss[tid + w];
      int i1 = si[tid], i2 = si[tid + w];
      if (m2 > m1 || (m2 == m1 && i2 < i1)) {
        sm[tid] = m2; ss[tid] = s2 + s1 * __expf(m1 - m2); si[tid] = i2;
      } else {
        ss[tid] = s1 + s2 * __expf(m2 - m1);
      }
    }
    __syncthreads();
  }
  if (tid == 0) {
    int idx = si[0];
    float ysoft = 1.0f / ss[0];
    int valid = done[b] ? 0 : 1;
    idx_out[b] = idx;
    if (valid)
      out[((size_t)b * Tsz + t) * Vsz + idx] = (1.0f - ysoft) + ysoft;
    out[(size_t)Bsz * Tsz * Vsz + (size_t)b * Tsz + t] = valid ? 1.0f : 0.0f;
  }
}

// commit generated token: embeds/K/V row update + eos/done advance
__global__ void __launch_bounds__(256)
k_commit(float* embeds, float* Kc, float* Vc, const float* E,
         const float* Wk, const float* Wv, int* eos, int* done, const int* idx) {
  __shared__ float row[Hsz];
  int b = blockIdx.x, tid = threadIdx.x;
  int e = eos[b], dn = done[b], id = idx[b];
  if (!dn) {
    size_t base = ((size_t)b * Ssz + e) * Hsz;
    for (int h = tid; h < Hsz; h += 256) {
      float v = E[(size_t)id * Hsz + h];
      row[h] = v;
      embeds[base + h] = v;
    }
    __syncthreads();
    for (int h = tid; h < Hsz; h += 256) {
      float kk = 0.0f, vv = 0.0f;
      for (int j = 0; j < Hsz; ++j) {
        float r = row[j];
        kk = fmaf(r, Wk[(size_t)j * Hsz + h], kk);
        vv = fmaf(r, Wv[(size_t)j * Hsz + h], vv);
      }
      Kc[base + h] = kk;
      Vc[base + h] = vv;
    }
  }
  __syncthreads();
  if (tid == 0) {
    eos[b] = e + (dn ? 0 : 1);
    done[b] = dn | (id == SEP_ID);
  }
}

// ---------------- host ----------------

extern "C" void kernel_launch(void* const* d_in, const int* in_sizes, int n_in,
                              void* d_out, int out_size, void* d_ws, size_t ws_size,
                              hipStream_t stream) {
  (void)in_sizes; (void)n_in; (void)ws_size;
  const int*   input_ids    = (const int*)d_in[0];
  const int*   eos_location = (const int*)d_in[1];
  const float* E     = (const float*)d_in[2];
  const float* Wq    = (const float*)d_in[3];
  const float* Wk    = (const float*)d_in[4];
  const float* Wv    = (const float*)d_in[5];
  const float* gamma = (const float*)d_in[6];
  const float* beta  = (const float*)d_in[7];
  const float* u_noise = (const float*)d_in[8];
  float* out = (float*)d_out;

  // workspace layout (floats)
  float* ws = (float*)d_ws;
  size_t o = 0;
  float* embeds = ws + o; o += (size_t)Bsz * Ssz * Hsz;
  float* Kc     = ws + o; o += (size_t)Bsz * Ssz * Hsz;
  float* Vc     = ws + o; o += (size_t)Bsz * Ssz * Hsz;
  float* bias   = ws + o; o += (size_t)Bsz * Ssz;
  float* mk     = ws + o; o += Hsz;
  float* mv     = ws + o; o += Hsz;
  float* mq     = ws + o; o += Hsz;
  float* h_eos  = ws + o; o += (size_t)Bsz * Hsz;
  float* logits = ws + o; o += (size_t)Bsz * Vsz;
  int* eos  = (int*)(ws + o); o += Bsz;
  int* done = (int*)(ws + o); o += Bsz;
  int* idx  = (int*)(ws + o); o += Bsz;

  long nout = (long)out_size;
  k_zero<<<(unsigned)((nout + 255) / 256), 256, 0, stream>>>(out, nout);
  k_init_state<<<1, Bsz, 0, stream>>>(eos_location, eos, done);
  k_bias<<<(Bsz * Ssz + 255) / 256, 256, 0, stream>>>(input_ids, bias);
  {
    long n = (long)Bsz * Ssz * Hsz;
    k_gather_embeds<<<(unsigned)((n + 255) / 256), 256, 0, stream>>>(input_ids, E, embeds);
  }
  k_mask_rows<<<(3 * Hsz + 255) / 256, 256, 0, stream>>>(E, Wk, Wv, Wq, mk, mv, mq);
  // K/V caches: [4096,768] @ [768,768] via f32 WMMA
  {
    dim3 g(Bsz * Ssz / 32, Hsz / 16);
    k_gemm_nn_wmma<<<g, 64, 0, stream>>>(embeds, Wk, Kc);
    k_gemm_nn_wmma<<<g, 64, 0, stream>>>(embeds, Wv, Vc);
  }

  const int NT = (Vsz + 15) / 16;  // 1908
  for (int t = 0; t < Tsz; ++t) {
    k_scatter_mask<<<Bsz, 256, 0, stream>>>(embeds, Kc, Vc, E, mk, mv, eos);
    k_attention<<<Bsz, 256, 0, stream>>>(embeds, Kc, Vc, bias, mq, gamma, beta, eos, h_eos);
    k_logits_wmma<<<NT, 64, 0, stream>>>(h_eos, E, logits);
    k_gumbel_argmax<<<Bsz, 256, 0, stream>>>(logits, u_noise, t, done, idx, out);
    k_commit<<<Bsz, 256, 0, stream>>>(embeds, Kc, Vc, E, Wk, Wv, eos, done, idx);
  }
}